// TripletLoss_21990232555841
// MI455X (gfx1250) — compile-verified
//
#include <hip/hip_runtime.h>
#include <math.h>

// ---------------------------------------------------------------------------
// TripletLoss on MI455X (gfx1250, wave32)
//   - Gram matrix via V_WMMA_F32_16X16X4_F32 (fp32, matches reference dtype)
//   - distance matrix never materialized: mining fused into GEMM epilogue
//   - mask @ inputs replaced by per-class sums (O(N*D) instead of O(N^2*D))
// ---------------------------------------------------------------------------

typedef __attribute__((ext_vector_type(2))) float v2f;
typedef __attribute__((ext_vector_type(8))) float v8f;

#define NROWS 4096
#define DIM   512
#define NCLS  512
#define MARGIN_F 0.3f

// ---------------- init: mining accumulators ----------------
__global__ __launch_bounds__(256) void k_init(unsigned* __restrict__ ap,
                                              unsigned* __restrict__ an) {
  int i = blockIdx.x * 256 + threadIdx.x;
  if (i < NROWS) {
    ap[i] = 0u;           // 0.0f  (all distances > 0, uint order == float order)
    an[i] = 0x7F800000u;  // +inf
  }
}

// ---------------- squared row norms: one wave per row ----------------
__global__ __launch_bounds__(256) void k_sq(const float* __restrict__ X,
                                            float* __restrict__ sq) {
  int w    = (int)(blockIdx.x * 256 + threadIdx.x) >> 5;  // row
  int lane = threadIdx.x & 31;
  const float* row = X + (size_t)w * DIM;
  float s = 0.f;
  for (int d = lane; d < DIM; d += 32) { float v = row[d]; s += v * v; }
  #pragma unroll
  for (int m = 16; m >= 1; m >>= 1) s += __shfl_xor(s, m, 32);
  if (lane == 0) sq[w] = s;
}

// ---------------- per-class sums (deterministic gather) ----------------
// one block per class; targets staged in LDS (16KB)
__global__ __launch_bounds__(256) void k_classsum(const float* __restrict__ X,
                                                  const int* __restrict__ tg,
                                                  float* __restrict__ csum,
                                                  float* __restrict__ ccnt) {
  __shared__ int t_s[NROWS];
  int c = blockIdx.x;
  for (int i = threadIdx.x; i < NROWS; i += 256) t_s[i] = tg[i];
  __syncthreads();
  int d0 = threadIdx.x, d1 = threadIdx.x + 256;
  float a0 = 0.f, a1 = 0.f;
  int cnt = 0;
  for (int i = 0; i < NROWS; ++i) {
    if (t_s[i] == c) {
      const float* r = X + (size_t)i * DIM;
      a0 += r[d0];
      a1 += r[d1];
      ++cnt;
    }
  }
  csum[(size_t)c * DIM + d0] = a0;
  csum[(size_t)c * DIM + d1] = a1;
  if (threadIdx.x == 0) ccnt[c] = (float)cnt;
}

// ---------------- total sum over all rows = sum of class sums ----------------
__global__ __launch_bounds__(256) void k_total(const float* __restrict__ csum,
                                               float* __restrict__ total) {
  for (int d = threadIdx.x; d < DIM; d += 256) {
    float t = 0.f;
    for (int c = 0; c < NCLS; ++c) t += csum[(size_t)c * DIM + d];
    total[d] = t;
  }
}

// ---------------- fused Gram (fp32 WMMA) + distance + hard mining ----------
// grid: (4096/128)^2 = 1024 blocks, 256 threads (8 waves)
// wave computes a 16x128 strip: 8 accumulators of 16x16, K-loop over 512 in
// steps of 4 (V_WMMA_F32_16X16X4_F32).
__global__ __launch_bounds__(256) void k_gram(const float* __restrict__ X,
                                              const float* __restrict__ sq,
                                              const int* __restrict__ tg,
                                              unsigned* __restrict__ ap,
                                              unsigned* __restrict__ an) {
  const int tile_row = (int)blockIdx.x / (NROWS / 128);
  const int tile_col = (int)blockIdx.x % (NROWS / 128);
  const int wave = threadIdx.x >> 5;
  const int lane = threadIdx.x & 31;
  const int hh   = lane >> 4;    // half-wave
  const int lidx = lane & 15;
  const int row_base = tile_row * 128 + wave * 16;
  const int col_base = tile_col * 128;

  // A fragment (16x4 fp32): lane holds X[row_base+lidx][k + 2*hh + {0,1}]
  // B fragment (4x16 fp32): lane holds X[col_base+lidx][k + 2*hh + {0,1}]
  // -> identical float2 addressing on row-major X (Gram => B = X rows too)
  const float* aptr = X + (size_t)(row_base + lidx) * DIM + 2 * hh;
  const float* bptr = X + (size_t)(col_base + lidx) * DIM + 2 * hh;

  v8f acc[8] = {};

  for (int k = 0; k < DIM; k += 4) {
    v2f av = *(const v2f*)(aptr + k);
    #pragma unroll
    for (int a = 0; a < 8; ++a) {
      v2f bv = *(const v2f*)(bptr + (size_t)a * 16 * DIM + k);
      acc[a] = __builtin_amdgcn_wmma_f32_16x16x4_f32(
          /*neg_a=*/false, av, /*neg_b=*/false, bv,
          /*c_mod=*/(short)0, acc[a], /*reuse_a=*/false, /*reuse_b=*/false);
    }
  }

  // ---- epilogue: dist = sqrt(max(sq_i + sq_j - 2 G, 1e-12)), mine top/bot ----
  // C layout: VGPR v, lanes 0-15 -> (M=v, N=lane); lanes 16-31 -> (M=v+8, N=lane-16)
  float sqrow[8];
  int   trow[8];
  #pragma unroll
  for (int v = 0; v < 8; ++v) {
    int r = row_base + v + 8 * hh;
    sqrow[v] = sq[r];
    trow[v]  = tg[r];
  }
  float sqc[8];
  int   tcol[8];
  #pragma unroll
  for (int a = 0; a < 8; ++a) {
    int cidx = col_base + 16 * a + lidx;
    sqc[a]  = sq[cidx];
    tcol[a] = tg[cidx];
  }

  const float INF = __uint_as_float(0x7F800000u);
  #pragma unroll
  for (int v = 0; v < 8; ++v) {
    float pmax = -INF;  // hardest positive (max dist, same class)
    float nmin =  INF;  // hardest negative (min dist, diff class)
    #pragma unroll
    for (int a = 0; a < 8; ++a) {
      float g  = acc[a][v];
      float d2 = sqrow[v] + sqc[a] - 2.0f * g;
      float dd = sqrtf(fmaxf(d2, 1e-12f));
      if (trow[v] == tcol[a]) pmax = fmaxf(pmax, dd);
      else                    nmin = fminf(nmin, dd);
    }
    // reduce across the 16 lanes that share this row (width-16 xor shuffle)
    #pragma unroll
    for (int m = 8; m >= 1; m >>= 1) {
      pmax = fmaxf(pmax, __shfl_xor(pmax, m, 16));
      nmin = fminf(nmin, __shfl_xor(nmin, m, 16));
    }
    if (lidx == 0) {
      int r = row_base + v + 8 * hh;
      if (pmax > 0.f) atomicMax(&ap[r], __float_as_uint(pmax));
      if (nmin < INF) atomicMin(&an[r], __float_as_uint(nmin));
    }
  }
}

// ---------------- per-row loss terms: one wave per row ----------------
__global__ __launch_bounds__(256) void k_rows(const float* __restrict__ X,
                                              const int* __restrict__ tg,
                                              const float* __restrict__ csum,
                                              const float* __restrict__ ccnt,
                                              const float* __restrict__ total,
                                              const float* __restrict__ apf,
                                              const float* __restrict__ anf,
                                              float* __restrict__ r0,
                                              float* __restrict__ r1,
                                              float* __restrict__ r2) {
  int w    = (int)(blockIdx.x * 256 + threadIdx.x) >> 5;  // row
  int lane = threadIdx.x & 31;
  int t = tg[w];
  float cnt   = ccnt[t];
  float inv_p = 1.0f / cnt;
  float inv_n = 1.0f / ((float)NROWS - cnt);
  const float* xr = X + (size_t)w * DIM;
  const float* cs = csum + (size_t)t * DIM;
  float s_cp = 0.f, s_cn = 0.f, s_cc = 0.f;
  for (int d = lane; d < DIM; d += 32) {
    float x   = xr[d];
    float csd = cs[d];
    float tot = total[d];
    float ic  = csd * inv_p;
    float oc  = (tot - csd) * inv_n;
    float e1 = ic - x, e2 = oc - x, e3 = ic - oc;
    s_cp += e1 * e1;
    s_cn += e2 * e2;
    s_cc += e3 * e3;
  }
  #pragma unroll
  for (int m = 16; m >= 1; m >>= 1) {
    s_cp += __shfl_xor(s_cp, m, 32);
    s_cn += __shfl_xor(s_cn, m, 32);
    s_cc += __shfl_xor(s_cc, m, 32);
  }
  if (lane == 0) {
    float dap = apf[w], dan = anf[w];
    r0[w] = fmaxf(dap - dan + MARGIN_F, 0.f);
    r1[w] = fmaxf(sqrtf(s_cp) - sqrtf(s_cn) + MARGIN_F, 0.f);
    r2[w] = sqrtf(s_cc);
  }
}

// ---------------- deterministic final reduction ----------------
__global__ __launch_bounds__(256) void k_reduce(const float* __restrict__ r0,
                                                const float* __restrict__ r1,
                                                const float* __restrict__ r2,
                                                float* __restrict__ out) {
  __shared__ float s0[256], s1[256], s2[256];
  float a0 = 0.f, a1 = 0.f, a2 = 0.f;
  for (int i = threadIdx.x; i < NROWS; i += 256) {
    a0 += r0[i];
    a1 += r1[i];
    a2 += r2[i];
  }
  s0[threadIdx.x] = a0;
  s1[threadIdx.x] = a1;
  s2[threadIdx.x] = a2;
  __syncthreads();
  for (int s = 128; s > 0; s >>= 1) {
    if (threadIdx.x < (unsigned)s) {
      s0[threadIdx.x] += s0[threadIdx.x + s];
      s1[threadIdx.x] += s1[threadIdx.x + s];
      s2[threadIdx.x] += s2[threadIdx.x + s];
    }
    __syncthreads();
  }
  if (threadIdx.x == 0) {
    const float invN = 1.0f / (float)NROWS;
    out[0] = s0[0] * invN;   // triplet_loss
    out[1] = s1[0] * invN;   // centroid_triplet_loss
    out[2] = -s2[0] * invN;  // cc_loss
  }
}

extern "C" void kernel_launch(void* const* d_in, const int* in_sizes, int n_in,
                              void* d_out, int out_size, void* d_ws, size_t ws_size,
                              hipStream_t stream) {
  const float* X  = (const float*)d_in[0];  // [4096, 512] f32
  const int*   tg = (const int*)d_in[1];    // [4096] int32 (jax int64 -> i32 w/o x64)
  float* out = (float*)d_out;               // 3 scalars

  // workspace layout (floats)
  float* ws = (float*)d_ws;
  float*    sq    = ws;                   // 4096
  unsigned* ap    = (unsigned*)(ws + 4096);  // 4096 (float bits)
  unsigned* an    = (unsigned*)(ws + 8192);  // 4096 (float bits)
  float*    r0    = ws + 12288;           // 4096
  float*    r1    = ws + 16384;           // 4096
  float*    r2    = ws + 20480;           // 4096
  float*    ccnt  = ws + 24576;           // 512
  float*    total = ws + 25088;           // 512
  float*    csum  = ws + 25600;           // 512*512

  k_init<<<NROWS / 256, 256, 0, stream>>>(ap, an);
  k_sq<<<NROWS / 8, 256, 0, stream>>>(X, sq);            // one wave per row
  k_classsum<<<NCLS, 256, 0, stream>>>(X, tg, csum, ccnt);
  k_total<<<1, 256, 0, stream>>>(csum, total);
  k_gram<<<(NROWS / 128) * (NROWS / 128), 256, 0, stream>>>(X, sq, tg, ap, an);
  k_rows<<<NROWS / 8, 256, 0, stream>>>(X, tg, csum, ccnt, total,
                                        (const float*)ap, (const float*)an,
                                        r0, r1, r2);
  k_reduce<<<1, 256, 0, stream>>>(r0, r1, r2, out);
}